// ParallelPairConstancyAxiom_40570261078451
// MI455X (gfx1250) — compile-verified
//
#include <hip/hip_runtime.h>
#include <math.h>

#define PI_F 3.14159265358979323846f
#define TH_F 0.08726646259971647f  /* radians(5.0) */

typedef float v2f __attribute__((ext_vector_type(2)));
typedef float v8f __attribute__((ext_vector_type(8)));

// ---------------- zero scratch ----------------
__global__ void zero_u32_kernel(unsigned* p, int n) {
    int i = blockIdx.x * blockDim.x + threadIdx.x;
    if (i < n) p[i] = 0u;
}

// ---------------- per-edge prep ----------------
__global__ void edge_prep_kernel(const float* __restrict__ pos,
                                 const int* __restrict__ eidx,
                                 int E, int nNodes,
                                 float* __restrict__ nx, float* __restrict__ ny,
                                 float* __restrict__ mx, float* __restrict__ my,
                                 float* __restrict__ cc, float* __restrict__ ang) {
    int e = blockIdx.x * blockDim.x + threadIdx.x;
    if (e >= E) return;
    int s = eidx[e];
    int d = eidx[E + e];
    s = min(max(s, 0), nNodes - 1);
    d = min(max(d, 0), nNodes - 1);
    float sx = pos[2 * s], sy = pos[2 * s + 1];
    float dx = pos[2 * d], dy = pos[2 * d + 1];
    float vx = dx - sx, vy = dy - sy;
    float len = fmaxf(sqrtf(vx * vx + vy * vy), 1e-8f);
    float ux = vx / len, uy = vy / len;
    float a = atan2f(uy, ux);
    a = fmodf(a, PI_F);
    if (a < 0.f) a += PI_F;
    ang[e] = a;
    float nxe = -uy, nye = ux;
    nx[e] = nxe;
    ny[e] = nye;
    mx[e] = 0.5f * (sx + dx);
    my[e] = 0.5f * (sy + dy);
    cc[e] = sx * nxe + sy * nye;
}

// ---------------- pairwise pass (WMMA tile sweep), mode specialized ----------------
// MODE 1: count n + histogram of key[31:21]   (ghist = 2048 bins)
// MODE 2: per-target histogram of key[20:10]  (ghist = 3*2048 bins)
// MODE 3: per-target histogram of key[9:0]    (ghist = 3*1024 bins)
// MODE 4: hinge-loss partial sums per block
template <int MODE>
__global__ __launch_bounds__(256)
void pair_pass_kernel(const float* __restrict__ nx, const float* __restrict__ ny,
                      const float* __restrict__ mx, const float* __restrict__ my,
                      const float* __restrict__ cc, const float* __restrict__ ang,
                      int E, int tilesPerDim,
                      const unsigned* __restrict__ sel,
                      const float* __restrict__ qparams,
                      unsigned* __restrict__ ghist,
                      unsigned* __restrict__ gcount,
                      float* __restrict__ partial) {
    constexpr int NBINS = (MODE == 1) ? 2048 : (MODE == 2) ? 6144 : (MODE == 3) ? 3072 : 1;
    __shared__ unsigned lhist[NBINS];
    __shared__ float lred[256];
    __shared__ unsigned lcnt;

    int tid = threadIdx.x;
    if (MODE <= 3) {
        for (int b = tid; b < NBINS; b += 256) lhist[b] = 0u;
    }
    if (MODE == 1 && tid == 0) lcnt = 0u;
    __syncthreads();

    int wave = tid >> 5;
    int lane = tid & 31;
    long tileId = (long)blockIdx.x * 8 + wave;
    long totalTiles = (long)tilesPerDim * (long)tilesPerDim;

    float s_loss = 0.f;

    // per-mode selection state (compile-time gated)
    unsigned p1[3] = {0u, 0u, 0u}, p2[3] = {0u, 0u, 0u};
    float mu = 0.f, margin = 0.f;
    if (MODE == 2) { p1[0] = sel[0]; p1[1] = sel[1]; p1[2] = sel[2]; }
    if (MODE == 3) { p2[0] = sel[6]; p2[1] = sel[7]; p2[2] = sel[8]; }
    if (MODE == 4) { mu = qparams[0]; margin = qparams[1]; }

    bool active = tileId < totalTiles;
    int ti = active ? (int)(tileId / tilesPerDim) : 0;
    int tj = active ? (int)(tileId % tilesPerDim) : 0;
    int i0 = ti * 16, j0 = tj * 16;
    bool doTile = active && (i0 <= j0 + 14);   // wave-uniform; tile intersects i<j

    if (doTile) {
        // A fragment: rows i (normals), K=0..1 used, K=2..3 zero
        int li = min(i0 + (lane & 15), E - 1);
        int lj = min(j0 + (lane & 15), E - 1);
        float anx = nx[li], anyv = ny[li];
        float bmx = mx[lj], bmy = my[lj];
        bool lo = lane < 16;
        v2f A, B;
        A.x = lo ? anx : 0.f;
        A.y = lo ? anyv : 0.f;
        B.x = lo ? bmx : 0.f;
        B.y = lo ? bmy : 0.f;
        v8f Cacc = {0.f, 0.f, 0.f, 0.f, 0.f, 0.f, 0.f, 0.f};
        // D[m][n] = normal_i . mid_j  via V_WMMA_F32_16X16X4_F32
        v8f dotv = __builtin_amdgcn_wmma_f32_16x16x4_f32(
            false, A, false, B, (short)0, Cacc, false, false);

        int hi = lane >> 4;              // 0 or 1
        int j = j0 + (lane & 15);
        float aj = ang[min(j, E - 1)];

#pragma unroll
        for (int r = 0; r < 8; ++r) {
            int m = r + hi * 8;
            int i = i0 + m;
            int ic = min(i, E - 1);
            float ai = ang[ic];
            float ci = cc[ic];
            float da = fabsf(ai - aj);
            float circ = fminf(da, PI_F - da);
            bool msk = (i < j) && (j < E) && (circ <= TH_F);
            float dist = fabsf(dotv[r] - ci);
            unsigned key = __float_as_uint(dist);   // monotonic for dist >= 0

            if (MODE == 1) {
                if (msk) {
                    atomicAdd(&lhist[key >> 21], 1u);
                    atomicAdd(&lcnt, 1u);
                }
            } else if (MODE == 2) {
                if (msk) {
                    unsigned top = key >> 21;
#pragma unroll
                    for (int t = 0; t < 3; ++t)
                        if (top == p1[t])
                            atomicAdd(&lhist[t * 2048 + ((key >> 10) & 0x7FFu)], 1u);
                }
            } else if (MODE == 3) {
                if (msk) {
                    unsigned top22 = key >> 10;
#pragma unroll
                    for (int t = 0; t < 3; ++t)
                        if (top22 == p2[t])
                            atomicAdd(&lhist[t * 1024 + (key & 0x3FFu)], 1u);
                }
            } else { // MODE 4: hinge loss
                if (msk) s_loss += fmaxf(fabsf(dist - mu) - margin, 0.f);
            }
        }
    }
    __syncthreads();

    if (MODE <= 3) {
        for (int b = tid; b < NBINS; b += 256) {
            unsigned v = lhist[b];
            if (v) atomicAdd(&ghist[b], v);
        }
        if (MODE == 1 && tid == 0 && lcnt) atomicAdd(gcount, lcnt);
    } else {
        lred[tid] = s_loss;
        __syncthreads();
        for (int st = 128; st > 0; st >>= 1) {
            if (tid < st) lred[tid] += lred[tid + st];
            __syncthreads();
        }
        if (tid == 0) partial[blockIdx.x] = lred[0];
    }
}

// ---------------- quantile selection (radix refine) ----------------
__global__ void select1_kernel(const unsigned* __restrict__ hist1,
                               const unsigned* __restrict__ gcount,
                               unsigned* __restrict__ sel) {
    if (blockIdx.x != 0 || threadIdx.x != 0) return;
    long n = (long)(*gcount);
    long k[3];
    k[0] = n / 4 - 1; if (k[0] < 0) k[0] = 0;
    k[1] = n / 2;
    k[2] = (3 * n) / 4;
    if (k[2] > n - 1) k[2] = n - 1;
    if (k[2] < 0) k[2] = 0;
    for (int t = 0; t < 3; ++t) {
        long cum = 0;
        unsigned pfx = 0;
        long rem = 0;
        for (int b = 0; b < 2048; ++b) {
            long c2 = cum + (long)hist1[b];
            if (k[t] < c2) { pfx = (unsigned)b; rem = k[t] - cum; break; }
            cum = c2;
        }
        sel[t] = pfx;
        sel[3 + t] = (unsigned)rem;
    }
}

__global__ void select2_kernel(const unsigned* __restrict__ hist2,
                               unsigned* __restrict__ sel) {
    if (blockIdx.x != 0 || threadIdx.x != 0) return;
    for (int t = 0; t < 3; ++t) {
        long k = (long)sel[3 + t];
        long cum = 0;
        unsigned pfx = sel[t] << 11;
        unsigned rem = 0;
        for (int b = 0; b < 2048; ++b) {
            long c2 = cum + (long)hist2[t * 2048 + b];
            if (k < c2) { pfx = (sel[t] << 11) | (unsigned)b; rem = (unsigned)(k - cum); break; }
            cum = c2;
        }
        sel[6 + t] = pfx;
        sel[9 + t] = rem;
    }
}

__global__ void select3_kernel(const unsigned* __restrict__ hist3,
                               const unsigned* __restrict__ sel,
                               float* __restrict__ qparams) {
    if (blockIdx.x != 0 || threadIdx.x != 0) return;
    float q[3] = {0.f, 0.f, 0.f};
    for (int t = 0; t < 3; ++t) {
        long k = (long)sel[9 + t];
        long cum = 0;
        for (int b = 0; b < 1024; ++b) {
            long c2 = cum + (long)hist3[t * 1024 + b];
            if (k < c2) {
                unsigned key = (sel[6 + t] << 10) | (unsigned)b;
                q[t] = __uint_as_float(key);
                break;
            }
            cum = c2;
        }
    }
    qparams[0] = q[1];                                   // mu (median)
    qparams[1] = fmaxf(q[2] - q[0], 1e-6f) * 0.75f;      // iqr * 0.5 * 1.5
}

// ---------------- deterministic final reduction ----------------
__global__ void final_reduce_kernel(const float* __restrict__ partial, int nPart,
                                    const unsigned* __restrict__ gcount,
                                    float* __restrict__ out) {
    __shared__ float lr[256];
    float s = 0.f;
    for (int i = threadIdx.x; i < nPart; i += 256) s += partial[i];
    lr[threadIdx.x] = s;
    __syncthreads();
    for (int st = 128; st > 0; st >>= 1) {
        if (threadIdx.x < st) lr[threadIdx.x] += lr[threadIdx.x + st];
        __syncthreads();
    }
    if (threadIdx.x == 0) {
        unsigned n = *gcount;
        out[0] = lr[0] / (float)(n > 0u ? n : 1u);
    }
}

extern "C" void kernel_launch(void* const* d_in, const int* in_sizes, int n_in,
                              void* d_out, int out_size, void* d_ws, size_t ws_size,
                              hipStream_t stream) {
    const float* pos = (const float*)d_in[0];        // (B*N, 2) flat
    const int* eidx = (const int*)d_in[2];           // (2, E) flat
    int E = in_sizes[2] / 2;
    int nNodes = in_sizes[0] / 2;

    float* ws = (float*)d_ws;
    float* nx = ws;
    float* ny = ws + (size_t)E;
    float* mx = ws + 2 * (size_t)E;
    float* my = ws + 3 * (size_t)E;
    float* cc = ws + 4 * (size_t)E;
    float* angp = ws + 5 * (size_t)E;
    unsigned* hist1 = (unsigned*)(ws + 6 * (size_t)E);
    unsigned* hist2 = hist1 + 2048;
    unsigned* hist3 = hist2 + 6144;
    unsigned* gcount = hist3 + 3072;
    unsigned* sel = gcount + 1;
    float* qparams = (float*)(sel + 12);
    float* partial = qparams + 2;

    int tilesPerDim = (E + 15) / 16;
    long totalTiles = (long)tilesPerDim * (long)tilesPerDim;
    int nBlocks = (int)((totalTiles + 7) / 8);

    int zeroWords = 2048 + 6144 + 3072 + 1 + 12 + 2;
    zero_u32_kernel<<<(zeroWords + 255) / 256, 256, 0, stream>>>(hist1, zeroWords);

    edge_prep_kernel<<<(E + 255) / 256, 256, 0, stream>>>(
        pos, eidx, E, nNodes, nx, ny, mx, my, cc, angp);

    // Pass 1: count + top-bit histogram
    pair_pass_kernel<1><<<nBlocks, 256, 0, stream>>>(
        nx, ny, mx, my, cc, angp, E, tilesPerDim, sel, qparams, hist1, gcount, partial);
    select1_kernel<<<1, 1, 0, stream>>>(hist1, gcount, sel);

    // Pass 2: middle-bit refinement per quantile target
    pair_pass_kernel<2><<<nBlocks, 256, 0, stream>>>(
        nx, ny, mx, my, cc, angp, E, tilesPerDim, sel, qparams, hist2, gcount, partial);
    select2_kernel<<<1, 1, 0, stream>>>(hist2, sel);

    // Pass 3: low-bit refinement -> exact q1, mu, q3
    pair_pass_kernel<3><<<nBlocks, 256, 0, stream>>>(
        nx, ny, mx, my, cc, angp, E, tilesPerDim, sel, qparams, hist3, gcount, partial);
    select3_kernel<<<1, 1, 0, stream>>>(hist3, sel, qparams);

    // Pass 4: hinge loss accumulation
    pair_pass_kernel<4><<<nBlocks, 256, 0, stream>>>(
        nx, ny, mx, my, cc, angp, E, tilesPerDim, sel, qparams, hist1, gcount, partial);
    final_reduce_kernel<<<1, 256, 0, stream>>>(partial, nBlocks, gcount, (float*)d_out);
}